// AutoEncoder_31121333027103
// MI455X (gfx1250) — compile-verified
//
#include <hip/hip_runtime.h>

typedef __attribute__((ext_vector_type(2))) float v2f;
typedef __attribute__((ext_vector_type(4))) float v4f;
typedef __attribute__((ext_vector_type(8))) float v8f;

#define HH 200
#define WW 200
#define KSZ 37          // kernel width (37x37, only <=36x36 slices used)
#define HALF 18
#define NP 64           // trajectory points per batch
#define MT 13           // 13*16 = 208 >= 200
#define NT 13
#define PAD 208         // padded leading dim for R, C, heat (13*16, mult of 4)
#define THREADS 256
#define NWAVE (THREADS / 32)

struct Smem {
  float heat[PAD * PAD];   // 208x208 accumulated image      (173056 B)
  float R[NP * PAD];       // [k][m] row profiles * 1/s^2     (53248 B)
  float C[NP * PAD];       // [k][n] col profiles             (53248 B)
  float pts[2 * NP];       // this batch's 64 (x,y) points
  float k1[40];            // 1-D gaussian taps
  float red[THREADS];      // max-reduction scratch
  float inv_s2;
  float scale;
};

__global__ __launch_bounds__(THREADS) void heatmap_splat_kernel(
    const float* __restrict__ x_t, float* __restrict__ out) {
  __shared__ Smem sm;
  const int tid = threadIdx.x;
  const int b = blockIdx.x;

  // ---- stage 0: gaussian taps, load points, zero R/C -----------------------
  if (tid < KSZ) {
    float d = (float)(tid - HALF);
    sm.k1[tid] = expf(-(d * d) * (1.0f / 18.0f));   // sigma = 3
  }
  if (tid < 2 * NP) sm.pts[tid] = x_t[b * (2 * NP) + tid];
  {
    v4f* z = (v4f*)sm.R;                            // R and C are contiguous
    const int n4 = (2 * NP * PAD) / 4;
    v4f zero4 = {0.f, 0.f, 0.f, 0.f};
    for (int i = tid; i < n4; i += THREADS) z[i] = zero4;
  }
  __syncthreads();
  if (tid == 0) {
    float s = 0.0f;
    for (int i = 0; i < KSZ; ++i) s += sm.k1[i];
    sm.inv_s2 = 1.0f / (s * s);                     // kernel normalization
  }
  __syncthreads();

  // ---- stage 1: build R (row profiles) and C (col profiles) ----------------
  const float inv_s2 = sm.inv_s2;
  for (int u = tid; u < 2 * NP * KSZ; u += THREADS) {
    int which = (u >= NP * KSZ);                    // 0 -> R, 1 -> C
    int v = which ? u - NP * KSZ : u;
    int t = v / KSZ;
    int idx = v - t * KSZ;
    float px = sm.pts[2 * t + 0];
    float py = sm.pts[2 * t + 1];
    if (px != px || py != py) continue;             // NaN -> invalid point
    if (!which) {
      int y_pix = HH - (int)(py * 2.0f);            // y_scale = 2
      int y_start = min(max(y_pix - HALF, 0), HH - (KSZ - 1));
      int y_end   = min(max(y_pix + HALF, 0), HH);
      if (idx < y_end - y_start)
        sm.R[t * PAD + y_start + idx] = sm.k1[idx] * inv_s2;
    } else {
      int x_pix = (int)(px * 2.0f);                 // x_scale = 2
      int x_start = min(max(x_pix - HALF, 0), WW - (KSZ - 1));
      int x_end   = min(max(x_pix + HALF, 0), WW);
      if (idx < x_end - x_start)
        sm.C[t * PAD + x_start + idx] = sm.k1[idx];
    }
  }
  __syncthreads();

  // ---- stage 2: heat = R^T * C  (M=N=208 padded, K=64) via f32 WMMA --------
  // Two independent 8-deep WMMA chains (even/odd K-chunks) to halve the
  // serial D->C dependency depth, summed at the end.
  const int wave = tid >> 5;
  const int lane = tid & 31;
  const int ll = lane & 15;               // M (for A) / N (for B) within tile
  const int kh = (lane >> 4) << 1;        // lane-half selects K pair {0,2}
  for (int tile = wave; tile < MT * NT; tile += NWAVE) {
    int tm = tile / NT;
    int tn = tile - tm * NT;
    const float* Rp = &sm.R[tm * 16 + ll];
    const float* Cp = &sm.C[tn * 16 + ll];
    v8f acc0 = {0.f, 0.f, 0.f, 0.f, 0.f, 0.f, 0.f, 0.f};
    v8f acc1 = {0.f, 0.f, 0.f, 0.f, 0.f, 0.f, 0.f, 0.f};
#pragma unroll
    for (int k0 = 0; k0 < NP; k0 += 8) {
      int ka = (k0 + kh) * PAD;
      int kb = (k0 + 4 + kh) * PAD;
      v2f a0, b0, a1, b1;
      a0.x = Rp[ka];                      // A[m][k0+kh]
      a0.y = Rp[ka + PAD];                // A[m][k0+kh+1]
      b0.x = Cp[ka];                      // B[k0+kh][n]
      b0.y = Cp[ka + PAD];                // B[k0+kh+1][n]
      a1.x = Rp[kb];
      a1.y = Rp[kb + PAD];
      b1.x = Cp[kb];
      b1.y = Cp[kb + PAD];
      acc0 = __builtin_amdgcn_wmma_f32_16x16x4_f32(
          false, a0, false, b0, (short)0, acc0, false, false);
      acc1 = __builtin_amdgcn_wmma_f32_16x16x4_f32(
          false, a1, false, b1, (short)0, acc1, false, false);
    }
    int row0 = tm * 16 + ((lane >> 4) << 3);   // C/D layout: +8 for hi lanes
    int col  = tn * 16 + ll;
#pragma unroll
    for (int v = 0; v < 8; ++v)
      sm.heat[(row0 + v) * PAD + col] = acc0[v] + acc1[v];
  }
  __syncthreads();

  // ---- stage 3: per-image max (vectorized LDS sweep) -----------------------
  {
    const v4f* hp = (const v4f*)sm.heat;
    float m = 0.0f;                        // heat >= 0 everywhere
    for (int i = tid; i < (PAD * PAD) / 4; i += THREADS) {
      v4f h = hp[i];
      m = fmaxf(m, fmaxf(fmaxf(h.x, h.y), fmaxf(h.z, h.w)));
    }
    sm.red[tid] = m;
  }
  __syncthreads();
  for (int s = THREADS / 2; s > 0; s >>= 1) {
    if (tid < s) sm.red[tid] = fmaxf(sm.red[tid], sm.red[tid + s]);
    __syncthreads();
  }
  if (tid == 0) sm.scale = 1.0f / (sm.red[0] + 1e-10f);
  __syncthreads();

  // ---- stage 4: normalized writeout, b128 LDS loads + b128 global stores ---
  // WW % 4 == 0, so a 4-aligned flat segment never crosses a row; LDS address
  // (y*PAD + x)*4 is 16B-aligned because PAD % 4 == 0 and x % 4 == 0.
  const float scale = sm.scale;
  v4f* outp4 = (v4f*)(out + (size_t)b * (HH * WW));
  for (int i4 = tid; i4 < (HH * WW) / 4; i4 += THREADS) {
    int f = i4 * 4;
    int y = f / WW;
    int x = f - y * WW;
    v4f h = *(const v4f*)&sm.heat[y * PAD + x];
    h.x *= scale; h.y *= scale; h.z *= scale; h.w *= scale;
    outp4[i4] = h;
  }
}

extern "C" void kernel_launch(void* const* d_in, const int* in_sizes, int n_in,
                              void* d_out, int out_size, void* d_ws, size_t ws_size,
                              hipStream_t stream) {
  const float* x_t = (const float*)d_in[0];
  float* out = (float*)d_out;
  const int B = in_sizes[0] / (2 * NP);   // 512 batches
  heatmap_splat_kernel<<<B, THREADS, 0, stream>>>(x_t, out);
}